// BP_Decoder_78099685310593
// MI455X (gfx1250) — compile-verified
//
#include <hip/hip_runtime.h>

// LDPC sum-product BP, damped, T iterations. See reference for semantics.
#define N_VARS   8192
#define N_CHKS   4096
#define DVD      3
#define N_EDGES  (N_VARS * DVD)     // 24576
#define BATCH_N  128
#define T_ITERS  30
#define BP_EPS   1e-7f
#define VPB      4                  // variables per block -> blockDim (128, 4) = 16 waves

// ---------------------------------------------------------------------------
// Phase A: variable -> check.  One thread per (variable v, batch b).
// Edges of v are contiguous: e = 3v+k.  Computes damped msg_V2C, then
// tanh/log-domain magnitude+sign and scatters onto check accumulators via
// L2-side f32 atomics (magnitude) and a u32 XOR atomic (sign parity).
// If CACHE_LA, also stores logabs so phase B skips tanhf+logf entirely.
// ---------------------------------------------------------------------------
template <bool CACHE_LA>
__global__ __launch_bounds__(BATCH_N * VPB)
void bp_var_kernel(const float* __restrict__ chn_llr,
                   const float* __restrict__ gamma_logit,
                   const int*   __restrict__ edge_chk,
                   const float* __restrict__ msgC2V,
                   float*       __restrict__ msgV2C,
                   float*       __restrict__ chk_log,
                   unsigned int* __restrict__ chk_par,
                   float*       __restrict__ la_cache) {
    const int b = threadIdx.x;                       // 0..127 (coalesced dim)
    const int v = blockIdx.x * VPB + threadIdx.y;    // variable node
    const float gamma = 1.0f / (1.0f + __expf(-gamma_logit[0]));  // sigmoid
    const float omg   = 1.0f - gamma;

    const int e0 = v * DVD;
    const size_t row0 = (size_t)e0 * BATCH_N + b;
    const float c0 = msgC2V[row0];
    const float c1 = msgC2V[row0 + BATCH_N];
    const float c2 = msgC2V[row0 + 2 * BATCH_N];
    const float base = chn_llr[(size_t)v * BATCH_N + b] + c0 + c1 + c2;

    #pragma unroll
    for (int k = 0; k < DVD; ++k) {
        const float ck  = (k == 0) ? c0 : ((k == 1) ? c1 : c2);
        const size_t ix = row0 + (size_t)k * BATCH_N;
        // damped variable->check message
        const float m = gamma * (base - ck) + omg * msgV2C[ix];
        msgV2C[ix] = m;
        // tanh product rule, log/sign domain
        const float mc  = fminf(fmaxf(m, -15.0f), 15.0f);
        const float t   = tanhf(0.5f * mc);
        const float la  = logf(fmaxf(fabsf(t), BP_EPS));
        if (CACHE_LA) la_cache[ix] = la;
        const int   chk = edge_chk[e0 + k];          // wave-uniform index
        const size_t cx = (size_t)chk * BATCH_N + b;
        atomicAdd(&chk_log[cx], la);                 // GLOBAL_ATOMIC_ADD_F32 (L2)
        if (t < 0.0f) atomicXor(&chk_par[cx], 1u);   // sign parity only
    }
}

// ---------------------------------------------------------------------------
// Phase B: check -> variable.  Gathers check totals, forms the extrinsic
// product, msg_C2V = 2*atanh = log((1+p)/(1-p)), and emits the per-iteration
// posterior with NON-TEMPORAL stores (write-once 126MB stream: keep the 192MB
// L2 for the message state, not dead output lines).
// CACHE_LA path: logabs comes from the L2-hot cache written in phase A; the
// sign is recovered exactly from msg_V2C (t<0 <=> m<0: clip & tanh preserve
// sign, and -0.0<0 is false on both sides) — no tanhf/logf recompute at all.
// ---------------------------------------------------------------------------
template <bool CACHE_LA>
__global__ __launch_bounds__(BATCH_N * VPB)
void bp_chk_kernel(const float* __restrict__ chn_llr,
                   const int*   __restrict__ edge_chk,
                   const float* __restrict__ msgV2C,
                   float*       __restrict__ msgC2V,
                   const float* __restrict__ chk_log,
                   const unsigned int* __restrict__ chk_par,
                   const float* __restrict__ la_cache,
                   float*       __restrict__ out_t) {
    const int b = threadIdx.x;
    const int v = blockIdx.x * VPB + threadIdx.y;
    const int e0 = v * DVD;
    const size_t row0 = (size_t)e0 * BATCH_N + b;

    float acc = 0.0f;
    #pragma unroll
    for (int k = 0; k < DVD; ++k) {
        const size_t ix = row0 + (size_t)k * BATCH_N;
        const float m = msgV2C[ix];
        float la;
        if (CACHE_LA) {
            la = la_cache[ix];
        } else {
            const float mc = fminf(fmaxf(m, -15.0f), 15.0f);
            const float t  = tanhf(0.5f * mc);
            la = logf(fmaxf(fabsf(t), BP_EPS));
        }
        const unsigned neg = (m < 0.0f) ? 1u : 0u;   // == (t < 0), exactly

        const int   chk = edge_chk[e0 + k];
        const size_t cx = (size_t)chk * BATCH_N + b;
        const float    slog = chk_log[cx];
        const unsigned spar = chk_par[cx];

        const float ext  = __expf(slog - la);                  // |prod| (exclude self)
        const float sign = ((spar ^ neg) & 1u) ? -1.0f : 1.0f; // mod-2 sign rule
        float prod = sign * ext;
        prod = fminf(fmaxf(prod, -1.0f + BP_EPS), 1.0f - BP_EPS);
        const float c2v = logf((1.0f + prod) / (1.0f - prod)); // 2*atanh(prod)
        msgC2V[ix] = c2v;
        acc += c2v;
    }
    const float outv = chn_llr[(size_t)v * BATCH_N + b] + acc;
    __builtin_nontemporal_store(outv, &out_t[(size_t)v * BATCH_N + b]); // TH=NT
}

// ---------------------------------------------------------------------------
// Host launcher.  Workspace layout (all L2-resident):
//   [msgC2V: E*B f32][msgV2C: E*B f32][chk_log: C*B f32][chk_par: C*B u32]
//   [la_cache: E*B f32]  (only if ws_size permits; 42MB total, else 29.4MB)
// Messages zeroed once per launch; check accumulators re-zeroed each
// iteration (contiguous -> single async memset node, graph-capture safe).
// ws_size is constant for a run, so the path choice is deterministic.
// ---------------------------------------------------------------------------
extern "C" void kernel_launch(void* const* d_in, const int* in_sizes, int n_in,
                              void* d_out, int out_size, void* d_ws, size_t ws_size,
                              hipStream_t stream) {
    (void)in_sizes; (void)n_in; (void)out_size;
    const float* chn_llr     = (const float*)d_in[0];
    const float* gamma_logit = (const float*)d_in[1];
    /* d_in[2] = edge_var: implied by layout (repeat(arange(N_VAR), DV)) */
    const int*   edge_chk    = (const int*)d_in[3];
    float* out = (float*)d_out;

    const size_t msg_elems = (size_t)N_EDGES * BATCH_N;   // 3,145,728
    const size_t chk_elems = (size_t)N_CHKS  * BATCH_N;   //   524,288
    float*        msgC2V  = (float*)d_ws;
    float*        msgV2C  = msgC2V + msg_elems;
    float*        chk_log = msgV2C + msg_elems;
    unsigned int* chk_par = (unsigned int*)(chk_log + chk_elems);
    float*        la_cache = (float*)(chk_par + chk_elems);

    const size_t need_full = (3 * msg_elems + 2 * chk_elems) * 4;
    const bool use_cache = (ws_size >= need_full);

    // init messages to zero (fresh state every call -> deterministic)
    hipMemsetAsync(msgC2V, 0, 2 * msg_elems * sizeof(float), stream);

    const dim3 blk(BATCH_N, VPB, 1);
    const dim3 grd(N_VARS / VPB, 1, 1);

    for (int t = 0; t < T_ITERS; ++t) {
        // zero chk_log + chk_par (adjacent -> one memset node)
        hipMemsetAsync(chk_log, 0, chk_elems * (sizeof(float) + sizeof(unsigned int)),
                       stream);
        float* out_t = out + (size_t)t * N_VARS * BATCH_N;
        if (use_cache) {
            bp_var_kernel<true><<<grd, blk, 0, stream>>>(chn_llr, gamma_logit, edge_chk,
                                                         msgC2V, msgV2C, chk_log,
                                                         chk_par, la_cache);
            bp_chk_kernel<true><<<grd, blk, 0, stream>>>(chn_llr, edge_chk, msgV2C,
                                                         msgC2V, chk_log, chk_par,
                                                         la_cache, out_t);
        } else {
            bp_var_kernel<false><<<grd, blk, 0, stream>>>(chn_llr, gamma_logit, edge_chk,
                                                          msgC2V, msgV2C, chk_log,
                                                          chk_par, nullptr);
            bp_chk_kernel<false><<<grd, blk, 0, stream>>>(chn_llr, edge_chk, msgV2C,
                                                          msgC2V, chk_log, chk_par,
                                                          nullptr, out_t);
        }
    }
}